// ACTModule_67070209294918
// MI455X (gfx1250) — compile-verified
//
#include <hip/hip_runtime.h>

// ---------------------------------------------------------------------------
// ACT module, restructured:
//   base      = inputs + time_enc                         [B*T, H]
//   baseT     = base @ Wt^T          (single fp32 WMMA GEMM, fused epilogue)
//   posT[k]   = pos_enc[k] @ Wt^T                          [12, H]
//   per-row:  s1=Σb, s2=Σb², dw=b·(γ⊙w_p), c_k=b·pos_k     (skinny reductions)
//   ACT loop over 12 hops entirely in per-token scalars -> W_k, S, rem, n_up
//   out_prev  = S*(baseT + bt) + Σ_k W_k * posT[k]
// ---------------------------------------------------------------------------

#define Bb 8
#define Tt 2048
#define Hh 1024
#define ROWS (Bb * Tt)            // 16384
#define HOPS 12
#define PREV_ELEMS (ROWS * Hh)    // 16777216
#define THRESH 0.9f
#define LN_EPS 1e-5f

typedef float v2f __attribute__((ext_vector_type(2)));
typedef float v8f __attribute__((ext_vector_type(8)));

// --------------------------- per-hop constants ------------------------------
// kc[k]=Σ pos_k, kc[12+k]=Σ pos_k², kc[24+k]=pos_k·(γ⊙w_p), kc[36]=Σγ⊙w_p, kc[37]=Σβ⊙w_p
__global__ __launch_bounds__(256)
void act_prep_consts(const float* __restrict__ pos_enc,
                     const float* __restrict__ gamma,
                     const float* __restrict__ beta,
                     const float* __restrict__ w_p,
                     float* __restrict__ kc) {
  __shared__ float red[256][3];
  const int tid = threadIdx.x, blk = blockIdx.x;
  float a = 0.f, b = 0.f, c = 0.f;
  if (blk == 0) {
    for (int h = tid; h < Hh; h += 256) {
      a += gamma[h] * w_p[h];
      b += beta[h] * w_p[h];
    }
  } else {
    const float* pr = pos_enc + (blk - 1) * Hh;
    for (int h = tid; h < Hh; h += 256) {
      float pv = pr[h];
      a += pv;
      b += pv * pv;
      c += pv * gamma[h] * w_p[h];
    }
  }
  red[tid][0] = a; red[tid][1] = b; red[tid][2] = c;
  __syncthreads();
  for (int s = 128; s > 0; s >>= 1) {
    if (tid < s) {
      red[tid][0] += red[tid + s][0];
      red[tid][1] += red[tid + s][1];
      red[tid][2] += red[tid + s][2];
    }
    __syncthreads();
  }
  if (tid == 0) {
    if (blk == 0) { kc[36] = red[0][0]; kc[37] = red[0][1]; }
    else {
      int k = blk - 1;
      kc[k] = red[0][0]; kc[12 + k] = red[0][1]; kc[24 + k] = red[0][2];
    }
  }
}

// --------------------------- posT = pos_enc @ Wt^T --------------------------
__global__ __launch_bounds__(256)
void act_posT(const float* __restrict__ pos_enc,
              const float* __restrict__ Wt,
              float* __restrict__ posT) {
  const int id = blockIdx.x * 256 + threadIdx.x;   // 12288 = 12*1024
  const int k = id >> 10, o = id & (Hh - 1);
  const float* pr = pos_enc + k * Hh;
  const float* wr = Wt + o * Hh;
  float s = 0.f;
  for (int h = 0; h < Hh; ++h) s += pr[h] * wr[h];
  posT[id] = s;   // id == k*H + o
}

// --------------------- per-row reductions (wave per row) --------------------
__global__ __launch_bounds__(256)
void act_row_stats(const float* __restrict__ inputs,
                   const float* __restrict__ time_enc,
                   const float* __restrict__ pos_enc,
                   const float* __restrict__ gamma,
                   const float* __restrict__ w_p,
                   float* __restrict__ s1o, float* __restrict__ s2o,
                   float* __restrict__ dwo, float* __restrict__ co) {
  const int row  = blockIdx.x * 8 + (threadIdx.x >> 5);
  const int lane = threadIdx.x & 31;
  const int t = row & (Tt - 1);
  const float* xr = inputs + (size_t)row * Hh;
  const float* tr = time_enc + (size_t)t * Hh;
  float s1 = 0.f, s2 = 0.f, dw = 0.f;
  float ck[HOPS];
#pragma unroll
  for (int k = 0; k < HOPS; ++k) ck[k] = 0.f;

  for (int h = lane; h < Hh; h += 32) {
    float bv = xr[h] + tr[h];
    s1 += bv;
    s2 += bv * bv;
    dw += bv * gamma[h] * w_p[h];
#pragma unroll
    for (int k = 0; k < HOPS; ++k) ck[k] += bv * pos_enc[k * Hh + h];
  }
  // wave32 butterfly reduction
  for (int off = 16; off > 0; off >>= 1) {
    s1 += __shfl_xor(s1, off);
    s2 += __shfl_xor(s2, off);
    dw += __shfl_xor(dw, off);
#pragma unroll
    for (int k = 0; k < HOPS; ++k) ck[k] += __shfl_xor(ck[k], off);
  }
  if (lane == 0) {
    s1o[row] = s1; s2o[row] = s2; dwo[row] = dw;
#pragma unroll
    for (int k = 0; k < HOPS; ++k) co[row * HOPS + k] = ck[k];
  }
}

// ------------------------- ACT halting scalar loop --------------------------
__global__ __launch_bounds__(256)
void act_scalars(const float* __restrict__ s1o, const float* __restrict__ s2o,
                 const float* __restrict__ dwo, const float* __restrict__ co,
                 const float* __restrict__ kc, const float* __restrict__ b_p,
                 float* __restrict__ Wsc, float* __restrict__ out) {
  const int m = blockIdx.x * 256 + threadIdx.x;  // 16384 exactly
  const float s1 = s1o[m], s2 = s2o[m], dw = dwo[m];
  const float sw = kc[36], bw = kc[37], bp0 = b_p[0];
  const float invH = 1.0f / (float)Hh;

  float hp = 0.f, rem = 0.f, nup = 0.f;
  float uw[HOPS];
#pragma unroll
  for (int k = 0; k < HOPS; ++k) {
    const float psum = kc[k], psq = kc[12 + k], pw = kc[24 + k];
    const float c_k = co[m * HOPS + k];
    const float mu  = (s1 + psum) * invH;
    const float ex2 = (s2 + 2.f * c_k + psq) * invH;
    const float var = ex2 - mu * mu;
    const float inv = rsqrtf(var + LN_EPS);
    const float logit = inv * ((dw + pw) - mu * sw) + bw + bp0;
    const float p = 1.0f / (1.0f + expf(-logit));
    // exact reference ordering
    float sr  = (hp < 1.0f) ? 1.f : 0.f;
    float acc = hp + p * sr;
    float nh  = ((acc > THRESH) ? 1.f : 0.f) * sr;
    sr        = ((acc <= THRESH) ? 1.f : 0.f) * sr;
    hp  = hp + p * sr;
    rem = rem + nh * (1.0f - hp);
    hp  = hp + nh * rem;
    nup = nup + sr + nh;
    uw[k] = p * sr + nh * rem;
  }
  // W_k = uw_k * prod_{j>k} (1-uw_j)
  float prod = 1.f;
#pragma unroll
  for (int k = HOPS - 1; k >= 0; --k) {
    Wsc[m * HOPS + k] = uw[k] * prod;
    prod *= (1.0f - uw[k]);
  }
  out[PREV_ELEMS + m] = rem;
  out[PREV_ELEMS + ROWS + m] = nup;
}

// ---------------- fp32 WMMA GEMM (base @ Wt^T) with fused epilogue ----------
// 64x128 macro-tile / 128 threads (4 waves); each wave: 4 M-tiles x 2 N-tiles.
// A fragments are reused across both N-tiles -> ~40% less LDS read traffic
// per v_wmma_f32_16x16x4_f32 than a 1-N-tile wave.
__global__ __launch_bounds__(128)
void act_gemm_epilogue(const float* __restrict__ inputs,
                       const float* __restrict__ time_enc,
                       const float* __restrict__ Wt,
                       const float* __restrict__ bt,
                       const float* __restrict__ Wsc,   // [ROWS,12]
                       const float* __restrict__ posT,  // [12,H]
                       float* __restrict__ out) {
  __shared__ float lds_a[64][33];    // padded: conflict-free column reads
  __shared__ float lds_b[32][128];
  __shared__ float lds_w[64][HOPS];

  const int tid  = threadIdx.x;
  const int lane = tid & 31;
  const int wv   = tid >> 5;         // 0..3 -> 32-wide N slice
  const int hi   = lane >> 4;        // lane half (K / M-offset select)
  const int lo   = lane & 15;        // M for A-frag, N for B/C-frag
  const int gm0  = blockIdx.x * 64;
  const int gnb  = blockIdx.y * 128;
  const int nb0  = wv * 32;          // wave's N base within macro-tile

  v8f acc[8];
#pragma unroll
  for (int i = 0; i < 8; ++i) acc[i] = (v8f){0.f,0.f,0.f,0.f,0.f,0.f,0.f,0.f};

  for (int kb = 0; kb < Hh; kb += 32) {
    __syncthreads();
    // stage A block 64x32: base = inputs + time_enc computed on load
#pragma unroll
    for (int r = 0; r < 16; ++r) {
      const int idx = tid + r * 128;        // 2048 elems
      const int m = idx >> 5, k = idx & 31;
      const int gm = gm0 + m;
      const int t = gm & (Tt - 1);
      lds_a[m][k] = inputs[(size_t)gm * Hh + kb + k] +
                    time_enc[(size_t)t * Hh + kb + k];
    }
    // stage B block 32x128: Bmat[k][n] = Wt[(gnb+n)*H + kb+k]
#pragma unroll
    for (int r = 0; r < 32; ++r) {
      const int idx = tid + r * 128;        // 4096 elems
      const int k = idx & 31, n = idx >> 5;
      lds_b[k][n] = Wt[(size_t)(gnb + n) * Hh + kb + k];
    }
    // prefetch next K block (global_prefetch_b8) while this block computes
    if (kb + 32 < Hh) {
      __builtin_prefetch(&inputs[(size_t)(gm0 + (tid >> 1)) * Hh + kb + 32 + (tid & 1) * 16], 0, 1);
      __builtin_prefetch(&Wt[(size_t)(gnb + tid) * Hh + kb + 32], 0, 1);
    }
    __syncthreads();

#pragma unroll
    for (int kk = 0; kk < 32; kk += 4) {
      const int kA = kk + 2 * hi;          // per-lane K base (ISA A/B layout)
      v2f bf0, bf1;
      bf0.x = lds_b[kA + 0][nb0 + lo];
      bf0.y = lds_b[kA + 1][nb0 + lo];
      bf1.x = lds_b[kA + 0][nb0 + 16 + lo];
      bf1.y = lds_b[kA + 1][nb0 + 16 + lo];
      v2f a0, a1, a2, a3;
      a0.x = lds_a[ 0 + lo][kA]; a0.y = lds_a[ 0 + lo][kA + 1];
      a1.x = lds_a[16 + lo][kA]; a1.y = lds_a[16 + lo][kA + 1];
      a2.x = lds_a[32 + lo][kA]; a2.y = lds_a[32 + lo][kA + 1];
      a3.x = lds_a[48 + lo][kA]; a3.y = lds_a[48 + lo][kA + 1];
      acc[0] = __builtin_amdgcn_wmma_f32_16x16x4_f32(false, a0, false, bf0, (short)0, acc[0], false, false);
      acc[1] = __builtin_amdgcn_wmma_f32_16x16x4_f32(false, a1, false, bf0, (short)0, acc[1], false, false);
      acc[2] = __builtin_amdgcn_wmma_f32_16x16x4_f32(false, a2, false, bf0, (short)0, acc[2], false, false);
      acc[3] = __builtin_amdgcn_wmma_f32_16x16x4_f32(false, a3, false, bf0, (short)0, acc[3], false, false);
      acc[4] = __builtin_amdgcn_wmma_f32_16x16x4_f32(false, a0, false, bf1, (short)0, acc[4], false, false);
      acc[5] = __builtin_amdgcn_wmma_f32_16x16x4_f32(false, a1, false, bf1, (short)0, acc[5], false, false);
      acc[6] = __builtin_amdgcn_wmma_f32_16x16x4_f32(false, a2, false, bf1, (short)0, acc[6], false, false);
      acc[7] = __builtin_amdgcn_wmma_f32_16x16x4_f32(false, a3, false, bf1, (short)0, acc[7], false, false);
    }
  }

  // ---- fused epilogue: out = S*(acc + bt) + W · posT ----
  __syncthreads();
#pragma unroll
  for (int r = 0; r < 6; ++r) {                 // 64*12 = 768 elems
    const int idx = tid + r * 128;
    const int m = idx / HOPS, k = idx % HOPS;
    lds_w[m][k] = Wsc[(size_t)(gm0 + m) * HOPS + k];
  }
  __syncthreads();

#pragma unroll
  for (int j = 0; j < 2; ++j) {                 // the wave's two N-tiles
    const int gn = gnb + nb0 + j * 16 + lo;
    float pt[HOPS];
#pragma unroll
    for (int k = 0; k < HOPS; ++k) pt[k] = posT[k * Hh + gn];
    const float btv = bt[gn];
#pragma unroll
    for (int i = 0; i < 4; ++i) {
#pragma unroll
      for (int r = 0; r < 8; ++r) {
        const int ml = 16 * i + 8 * hi + r;     // ISA C/D layout: M = r + 8*hi
        float S = 0.f, dot = 0.f;
#pragma unroll
        for (int k = 0; k < HOPS; ++k) {
          const float w = lds_w[ml][k];
          S += w;
          dot += w * pt[k];
        }
        out[(size_t)(gm0 + ml) * Hh + gn] = S * (acc[j * 4 + i][r] + btv) + dot;
      }
    }
  }
}

// ---------------------------------------------------------------------------
extern "C" void kernel_launch(void* const* d_in, const int* in_sizes, int n_in,
                              void* d_out, int out_size, void* d_ws, size_t ws_size,
                              hipStream_t stream) {
  const float* inputs   = (const float*)d_in[0];
  const float* time_enc = (const float*)d_in[1];
  const float* pos_enc  = (const float*)d_in[2];
  const float* gamma    = (const float*)d_in[3];
  const float* beta     = (const float*)d_in[4];
  const float* w_p      = (const float*)d_in[5];
  const float* b_p      = (const float*)d_in[6];
  const float* Wt       = (const float*)d_in[7];
  const float* bt       = (const float*)d_in[8];
  float* out = (float*)d_out;

  float* ws   = (float*)d_ws;
  float* s1   = ws;                      // 16384
  float* s2   = ws + 16384;              // 16384
  float* dw   = ws + 32768;              // 16384
  float* c    = ws + 49152;              // 16384*12
  float* Wsc  = ws + 245760;             // 16384*12
  float* posT = ws + 442368;             // 12*1024
  float* kc   = ws + 454656;             // 40

  act_prep_consts<<<13, 256, 0, stream>>>(pos_enc, gamma, beta, w_p, kc);
  act_posT<<<48, 256, 0, stream>>>(pos_enc, Wt, posT);
  act_row_stats<<<ROWS / 8, 256, 0, stream>>>(inputs, time_enc, pos_enc, gamma, w_p,
                                              s1, s2, dw, c);
  act_scalars<<<ROWS / 256, 256, 0, stream>>>(s1, s2, dw, c, kc, b_p, Wsc, out);
  act_gemm_epilogue<<<dim3(ROWS / 64, Hh / 128), 128, 0, stream>>>(
      inputs, time_enc, Wt, bt, Wsc, posT, out);
}